// RecurrentDNNC_83184926589180
// MI455X (gfx1250) — compile-verified
//
#include <hip/hip_runtime.h>
#include <hip/hip_bf16.h>

typedef __attribute__((ext_vector_type(2))) float v2f;
typedef __attribute__((ext_vector_type(4))) float v4f;
typedef __attribute__((ext_vector_type(8))) float v8f;

// ---------------------------------------------------------------------------
// Kernel A: c[t] = M * x[t] + v, where M = Wd[:, :2] @ W1  (2x4),
//           v = Wd[:, :2] @ b1 + bd                        (2,)
// Implemented with V_WMMA_F32_16X16X4_F32:
//   A (16x4)  = M padded with zero rows      (rows 0,1 = M)
//   B (4x16)  = x^T tile, 16 timesteps wide  (one float2 load per lane)
//   C (16x16) = v broadcast into rows 0,1
//   D rows 0,1 / lanes 0..15 = (c0, c1) for the 16 timesteps.
// ISA layouts used (CDNA5 ISA):
//   A 16x4 f32 : lane = row (l&15); VGPR v + lane-half h -> K = 2h+v
//   B 4x16 f32 : lane = col (l&15); VGPR v + lane-half h -> K = 2h+v
//   C/D 16x16  : VGPR r, lane l -> (M = r + 8*(l>=16), N = l&15)
// ---------------------------------------------------------------------------
__global__ __launch_bounds__(256) void dnnc_precompute_c(
    const float* __restrict__ x, const float* __restrict__ W1,
    const float* __restrict__ b1, const float* __restrict__ Wd,
    const float* __restrict__ bd, float* __restrict__ cbuf, int S)
{
    const int lane = threadIdx.x & 31;
    const int wave = threadIdx.x >> 5;
    const int wavesPerBlock = blockDim.x >> 5;
    const int gwave  = blockIdx.x * wavesPerBlock + wave;
    const int nwaves = gridDim.x * wavesPerBlock;
    const int r = lane & 15;         // row (A) / col (B) index
    const int h = lane >> 4;         // lane half selects K pair

    // Fold fc1 into the cell input map (uniform scalar math, hoisted).
    const float wd00 = Wd[0], wd01 = Wd[1], wd10 = Wd[4], wd11 = Wd[5];
    const int k0 = 2 * h, k1 = 2 * h + 1;
    const float w1_0k0 = W1[k0],     w1_1k0 = W1[4 + k0];
    const float w1_0k1 = W1[k1],     w1_1k1 = W1[4 + k1];
    const float M0k0 = wd00 * w1_0k0 + wd01 * w1_1k0;   // M[0][k0]
    const float M0k1 = wd00 * w1_0k1 + wd01 * w1_1k1;   // M[0][k1]
    const float M1k0 = wd10 * w1_0k0 + wd11 * w1_1k0;   // M[1][k0]
    const float M1k1 = wd10 * w1_0k1 + wd11 * w1_1k1;   // M[1][k1]
    const float v0 = wd00 * b1[0] + wd01 * b1[1] + bd[0];
    const float v1 = wd10 * b1[0] + wd11 * b1[1] + bd[1];

    // A operand: rows 0,1 carry M, rest zero.
    v2f av;
    av.x = (r == 0) ? M0k0 : (r == 1) ? M1k0 : 0.0f;
    av.y = (r == 0) ? M0k1 : (r == 1) ? M1k1 : 0.0f;

    // C operand: bias v broadcast across rows 0 and 1.
    v8f cm = {};
    cm[0] = (h == 0) ? v0 : 0.0f;   // VGPR0, lanes 0-15 -> row M=0
    cm[1] = (h == 0) ? v1 : 0.0f;   // VGPR1, lanes 0-15 -> row M=1

    const int numTiles = S >> 4;              // 16 timesteps per WMMA
    v2f* __restrict__ cv = (v2f*)cbuf;

    for (int tile = gwave; tile < numTiles; tile += nwaves) {
        const int t0 = tile << 4;
        // B operand: x^T tile; lane (h,r) loads x[t0+r][2h .. 2h+1].
        v2f bv = *(const v2f*)(x + (size_t)(t0 + r) * 4 + 2 * h);
        v8f d = __builtin_amdgcn_wmma_f32_16x16x4_f32(
            /*neg_a=*/false, av, /*neg_b=*/false, bv,
            /*c_mod=*/(short)0, cm, /*reuse_a=*/false, /*reuse_b=*/false);
        if (lane < 16) {                     // rows 0,1 live in d[0], d[1]
            v2f st; st.x = d[0]; st.y = d[1];
            cv[t0 + lane] = st;              // contiguous float2 per lane
        }
    }
}

// ---------------------------------------------------------------------------
// Kernel B: the inherently-serial ReLU recurrence + fused fc2/sigmoid.
//   y_t = relu(A2 * y_{t-1} + c_t),  A2 = Wd[:, 2:4]
//   out_t = sigmoid(W2 * y_t + b2)   (off the critical path -> free)
// Single lane; 8-step unroll; c is L2-resident (16 MB << 192 MB L2);
// b128 loads/stores + global_prefetch well ahead of the read cursor.
// ---------------------------------------------------------------------------
__global__ void dnnc_scan(const float* __restrict__ cbuf,
                          const float* __restrict__ Wd,
                          const float* __restrict__ W2,
                          const float* __restrict__ b2,
                          float* __restrict__ out, int S)
{
    if (threadIdx.x != 0) return;

    const float a00 = Wd[2], a01 = Wd[3], a10 = Wd[6], a11 = Wd[7];
    const float w00 = W2[0], w01 = W2[1], w10 = W2[2], w11 = W2[3];
    const float q0 = b2[0], q1 = b2[1];

    float y0 = 0.0f, y1 = 0.0f;
    const v4f* __restrict__ cb = (const v4f*)cbuf;   // 2 timesteps per v4f
    v4f* __restrict__ ob = (v4f*)out;

    for (int t = 0; t < S; t += 8) {
        const int base = t >> 1;

        // Prefetch ~2K timesteps ahead (speculative; OOB is dropped).
        __builtin_prefetch(cbuf + 2 * t + 4096, 0, 1);

        float cc[16];
        *(v4f*)(cc + 0)  = cb[base + 0];
        *(v4f*)(cc + 4)  = cb[base + 1];
        *(v4f*)(cc + 8)  = cb[base + 2];
        *(v4f*)(cc + 12) = cb[base + 3];

        float oo[16];
#pragma unroll
        for (int k = 0; k < 8; ++k) {
            // Critical path: fma -> fma -> max, per component.
            float z0 = __builtin_fmaf(a01, y1, cc[2 * k]);
            z0       = __builtin_fmaf(a00, y0, z0);
            float z1 = __builtin_fmaf(a11, y1, cc[2 * k + 1]);
            z1       = __builtin_fmaf(a10, y0, z1);
            y0 = fmaxf(z0, 0.0f);
            y1 = fmaxf(z1, 0.0f);

            // fc2 + sigmoid: does not feed back; fills latency slots.
            float o0 = __builtin_fmaf(w00, y0, __builtin_fmaf(w01, y1, q0));
            float o1 = __builtin_fmaf(w10, y0, __builtin_fmaf(w11, y1, q1));
            oo[2 * k]     = 1.0f / (1.0f + __expf(-o0));
            oo[2 * k + 1] = 1.0f / (1.0f + __expf(-o1));
        }

        ob[base + 0] = *(v4f*)(oo + 0);
        ob[base + 1] = *(v4f*)(oo + 4);
        ob[base + 2] = *(v4f*)(oo + 8);
        ob[base + 3] = *(v4f*)(oo + 12);
    }
}

// ---------------------------------------------------------------------------
extern "C" void kernel_launch(void* const* d_in, const int* in_sizes, int n_in,
                              void* d_out, int out_size, void* d_ws, size_t ws_size,
                              hipStream_t stream) {
    (void)n_in; (void)out_size; (void)ws_size;
    const float* x  = (const float*)d_in[0];  // (1, S, 4)
    const float* W1 = (const float*)d_in[1];  // (2, 4)
    const float* b1 = (const float*)d_in[2];  // (2,)
    const float* Wd = (const float*)d_in[3];  // (2, 4)
    const float* bd = (const float*)d_in[4];  // (2,)
    const float* W2 = (const float*)d_in[5];  // (2, 2)
    const float* b2 = (const float*)d_in[6];  // (2,)
    const int S = in_sizes[0] / 4;            // 2^21

    float* cbuf = (float*)d_ws;               // S * 2 floats = 16 MB scratch

    // Phase 1+2 fold: bandwidth-bound WMMA map x -> c (L2-resident output).
    dnnc_precompute_c<<<512, 256, 0, stream>>>(x, W1, b1, Wd, bd, cbuf, S);
    // Phase 3: serial scan + fused fc2/sigmoid, one lane, latency-optimized.
    dnnc_scan<<<1, 32, 0, stream>>>(cbuf, Wd, W2, b2, (float*)d_out, S);
}